// FastAPLoss_42417097016612
// MI455X (gfx1250) — compile-verified
//
#include <hip/hip_runtime.h>
#include <hip/hip_bf16.h>
#include <math.h>

typedef __attribute__((ext_vector_type(2))) float v2f;
typedef __attribute__((ext_vector_type(8))) float v8f;

#define N_EMB 2048
#define D_EMB 128
#define NUM_BINS 10
#define NUM_EDGES 11
#define HIST_MAX 4.0f
#define NUM_CLASSES 100

// ---------------------------------------------------------------------------
// Kernel 1: row-normalize embeddings; one wave (32 lanes) per row (D=128 -> 4
// floats/lane). Also store sq[i] = sum(emb_norm^2) in f32 exactly like the
// reference (it recomputes sq rather than assuming 1.0).
// ---------------------------------------------------------------------------
__global__ void __launch_bounds__(256)
fastap_normalize_kernel(const float* __restrict__ emb,
                        float* __restrict__ out,
                        float* __restrict__ sq) {
    const int wave = threadIdx.x >> 5;
    const int lane = threadIdx.x & 31;
    const int row  = blockIdx.x * 8 + wave;

    const float4 v = ((const float4*)(emb + row * D_EMB))[lane];
    float s = v.x * v.x + v.y * v.y + v.z * v.z + v.w * v.w;
#pragma unroll
    for (int off = 16; off > 0; off >>= 1) s += __shfl_xor(s, off, 32);

    const float inv = 1.0f / sqrtf(s);
    float4 o;
    o.x = v.x * inv; o.y = v.y * inv; o.z = v.z * inv; o.w = v.w * inv;
    ((float4*)(out + row * D_EMB))[lane] = o;

    float s2 = o.x * o.x + o.y * o.y + o.z * o.z + o.w * o.w;
#pragma unroll
    for (int off = 16; off > 0; off >>= 1) s2 += __shfl_xor(s2, off, 32);
    if (lane == 0) sq[row] = s2;
}

// ---------------------------------------------------------------------------
// Kernel 2: Gram matrix via V_WMMA_F32_16X16X4_F32 + fused triangular-pulse
// soft histogram. Block = 256 threads (8 waves) owns a 16-row anchor strip;
// each wave computes 16x16 tiles across the 2048 columns (stride 8) and
// scatters the (at most 2) nonzero pulse weights per pair into a per-row LDS
// histogram.
//
// f32 16x16x4 A layout: lanes 0-15 hold M=lane (K pair 0,1), lanes 16-31 hold
// M=lane-16 (K pair 2,3), VGPR0=K even, VGPR1=K odd. B (= A of the column
// block, since G = E*E^T) has the identical load pattern with N in place of M.
// C/D layout: VGPR r, lanes 0-15 -> (M=r, N=lane), lanes 16-31 -> (M=r+8,
// N=lane-16).
// ---------------------------------------------------------------------------
__global__ void __launch_bounds__(256)
fastap_hist_kernel(const float* __restrict__ emb,
                   const float* __restrict__ sq,
                   const int*   __restrict__ labels,
                   float* __restrict__ pos_hist,
                   float* __restrict__ neg_hist) {
    __shared__ float s_hist[16][2 * NUM_EDGES];  // [row][pos 0..10 | neg 11..21]
    __shared__ float s_sq[16];
    __shared__ int   s_lab[16];

    const int tid   = threadIdx.x;
    const int wave  = tid >> 5;
    const int lane  = tid & 31;
    const int rBase = blockIdx.x * 16;

    for (int i = tid; i < 16 * 2 * NUM_EDGES; i += 256) (&s_hist[0][0])[i] = 0.0f;
    if (tid < 16) {
        s_sq[tid]  = sq[rBase + tid];
        s_lab[tid] = labels[rBase + tid];
    }
    __syncthreads();

    const int lane15   = lane & 15;
    const int laneHalf = lane >> 4;  // selects K pair (A/B) and M half (C/D)

    // Hoist the A fragments for all 32 K-steps (64 floats/lane).
    const float* arow = emb + (rBase + lane15) * D_EMB + laneHalf * 2;
    v2f afrag[32];
#pragma unroll
    for (int k = 0; k < 32; ++k) afrag[k] = *(const v2f*)(arow + 4 * k);

    const float delta_inv = (float)NUM_BINS / HIST_MAX;  // 1/0.4

    for (int ct = wave; ct < N_EMB / 16; ct += 8) {
        const int cBase = ct * 16;
        const float* brow = emb + (cBase + lane15) * D_EMB + laneHalf * 2;

        v8f c = {};
#pragma unroll
        for (int k = 0; k < 32; ++k) {
            v2f b = *(const v2f*)(brow + 4 * k);
            c = __builtin_amdgcn_wmma_f32_16x16x4_f32(
                    /*neg_a=*/false, afrag[k],
                    /*neg_b=*/false, b,
                    /*c_mod=*/(short)0, c,
                    /*reuse_a=*/false, /*reuse_b=*/false);
        }

        const int   col  = cBase + lane15;
        const float sqj  = sq[col];
        const int   labj = labels[col];
        const int   rOff = laneHalf * 8;

#pragma unroll
        for (int r = 0; r < 8; ++r) {
            const int   rloc = r + rOff;
            const int   row  = rBase + rloc;
            const float d    = s_sq[rloc] + sqj - 2.0f * c[r];

            const bool same = (s_lab[rloc] == labj);
            const bool isPos = same && (row != col);
            const bool isNeg = !same;
            if (isPos || isNeg) {
                const int   base = isPos ? 0 : NUM_EDGES;
                const float t    = d * delta_inv;            // d / delta
                const int   e0   = (int)floorf(t);
                const float frac = t - (float)e0;
                // edge e0 gets (1-frac), edge e0+1 gets frac  ==  relu(1-|t-e|)
                if (e0 >= 0 && e0 < NUM_EDGES)
                    atomicAdd(&s_hist[rloc][base + e0], 1.0f - frac);
                const int e1 = e0 + 1;
                if (e1 >= 0 && e1 < NUM_EDGES && frac > 0.0f)
                    atomicAdd(&s_hist[rloc][base + e1], frac);
            }
        }
    }
    __syncthreads();

    for (int i = tid; i < 16 * NUM_EDGES; i += 256) {
        const int rloc = i / NUM_EDGES;
        const int e    = i % NUM_EDGES;
        pos_hist[(rBase + rloc) * NUM_EDGES + e] = s_hist[rloc][e];
        neg_hist[(rBase + rloc) * NUM_EDGES + e] = s_hist[rloc][NUM_EDGES + e];
    }
}

// ---------------------------------------------------------------------------
// Kernel 3: per-anchor FastAP (cumsum over 11 edges), N_pos via a 100-bin
// label-count histogram (labels in [0,100)), masked mean -> scalar loss.
// Single block; the count histogram replaces the O(N^2) label scan.
// ---------------------------------------------------------------------------
__global__ void __launch_bounds__(256)
fastap_reduce_kernel(const float* __restrict__ pos_hist,
                     const float* __restrict__ neg_hist,
                     const int*   __restrict__ labels,
                     float* __restrict__ out) {
    __shared__ int   s_cnt_lab[NUM_CLASSES];
    __shared__ float s_sum[256];
    __shared__ float s_cnt[256];

    const int tid = threadIdx.x;
    if (tid < NUM_CLASSES) s_cnt_lab[tid] = 0;
    __syncthreads();
    for (int i = tid; i < N_EMB; i += 256) atomicAdd(&s_cnt_lab[labels[i]], 1);
    __syncthreads();

    float sumv = 0.0f, cnt = 0.0f;
    for (int i = tid; i < N_EMB; i += 256) {
        const int npos = s_cnt_lab[labels[i]] - 1;  // exclude self

        float cum_pos = 0.0f, cum_tot = 0.0f, fap = 0.0f;
#pragma unroll
        for (int e = 0; e < NUM_EDGES; ++e) {
            const float ph = pos_hist[i * NUM_EDGES + e];
            const float nh = neg_hist[i * NUM_EDGES + e];
            cum_pos += ph;
            cum_tot += ph + nh;
            const float hp = ph * cum_pos;
            if (hp > 0.0f && cum_tot > 0.0f) fap += hp / cum_tot;
        }
        if (npos > 0) {
            sumv += 1.0f - fap / (float)npos;
            cnt  += 1.0f;
        }
    }
    s_sum[tid] = sumv;
    s_cnt[tid] = cnt;
    __syncthreads();
#pragma unroll
    for (int s = 128; s > 0; s >>= 1) {
        if (tid < s) {
            s_sum[tid] += s_sum[tid + s];
            s_cnt[tid] += s_cnt[tid + s];
        }
        __syncthreads();
    }
    if (tid == 0) out[0] = s_sum[0] / fmaxf(s_cnt[0], 1.0f);
}

// ---------------------------------------------------------------------------
extern "C" void kernel_launch(void* const* d_in, const int* in_sizes, int n_in,
                              void* d_out, int out_size, void* d_ws, size_t ws_size,
                              hipStream_t stream) {
    const float* emb    = (const float*)d_in[0];
    const int*   labels = (const int*)d_in[1];
    float*       out    = (float*)d_out;

    // Workspace layout (floats): norm[N*D] | sq[N] | pos_hist[N*11] | neg_hist[N*11]
    float* ws_norm = (float*)d_ws;
    float* ws_sq   = ws_norm + (size_t)N_EMB * D_EMB;
    float* ws_pos  = ws_sq + N_EMB;
    float* ws_neg  = ws_pos + (size_t)N_EMB * NUM_EDGES;

    fastap_normalize_kernel<<<N_EMB / 8, 256, 0, stream>>>(emb, ws_norm, ws_sq);
    fastap_hist_kernel<<<N_EMB / 16, 256, 0, stream>>>(ws_norm, ws_sq, labels,
                                                       ws_pos, ws_neg);
    fastap_reduce_kernel<<<1, 256, 0, stream>>>(ws_pos, ws_neg, labels, out);
}